// MaxMarginLoss_56203942035545
// MI455X (gfx1250) — compile-verified
//
#include <hip/hip_runtime.h>
#include <hip/hip_bf16.h>
#include <math.h>

typedef float v2f __attribute__((ext_vector_type(2)));
typedef float v8f __attribute__((ext_vector_type(8)));

#define BATCH 64
#define LQ 32
#define LD 256
#define DIM 128
#define MARGIN 0.2f

// ---------------------------------------------------------------------------
// Kernel 1: L2-normalize rows of q and d into workspace.
// One wave (32 lanes) per row of 128 floats; each lane handles a float4.
// ---------------------------------------------------------------------------
__global__ __launch_bounds__(256) void l2norm_kernel(const float* __restrict__ q,
                                                     const float* __restrict__ d,
                                                     float* __restrict__ qn,
                                                     float* __restrict__ dn) {
    const int nq_rows = BATCH * LQ;                 // 2048
    int row  = blockIdx.x * 8 + (threadIdx.x >> 5); // wave index = row index
    int lane = threadIdx.x & 31;

    const float* src;
    float* dst;
    if (row < nq_rows) {
        src = q  + (size_t)row * DIM;
        dst = qn + (size_t)row * DIM;
    } else {
        int r = row - nq_rows;
        src = d  + (size_t)r * DIM;
        dst = dn + (size_t)r * DIM;
    }

    float4 v = *(const float4*)(src + lane * 4);
    float ss = v.x * v.x + v.y * v.y + v.z * v.z + v.w * v.w;
#pragma unroll
    for (int s = 16; s >= 1; s >>= 1) ss += __shfl_xor(ss, s);
    float norm  = sqrtf(ss);
    float scale = 1.0f / fmaxf(norm, 1e-12f);
    v.x *= scale; v.y *= scale; v.z *= scale; v.w *= scale;
    *(float4*)(dst + lane * 4) = v;
}

// ---------------------------------------------------------------------------
// Kernel 2: per batch-pair (i,j) compute max over the 32x256 token-sim block
// via fp32 WMMA (16x16x4). Block = 512 threads = 16 waves; wave w owns doc
// column tile w (16 doc rows) and computes both query M-tiles (2 WMMA tiles).
//
// Fragment layout (V_WMMA_F32_16X16X4_F32, wave32):
//   A (16x4):  lane l<16 -> row M=l,    K={0,1} in V0,V1; l>=16 -> K={2,3}
//   B (4x16):  lane l<16 -> col N=l,    K={0,1} in V0,V1; l>=16 -> K={2,3}
//   C (16x16): VGPR v, lanes 0-15: M=v, N=lane; lanes 16-31: M=v+8, N=lane-16
// ---------------------------------------------------------------------------
__global__ __launch_bounds__(512) void maxsim_kernel(const float* __restrict__ qn,
                                                     const float* __restrict__ dn,
                                                     float* __restrict__ simmax) {
    const int i = blockIdx.x;   // query batch index
    const int j = blockIdx.y;   // doc batch index
    const int wave = threadIdx.x >> 5;  // 0..15 -> doc column tile
    const int lane = threadIdx.x & 31;
    const int lo = lane & 15;
    const int hi = lane >> 4;           // selects K pair {0,1} or {2,3}

    const float* qrow0 = qn + (size_t)i * (LQ * DIM) + (size_t)lo * DIM;         // M-tile 0
    const float* qrow1 = qrow0 + (size_t)16 * DIM;                                // M-tile 1
    const float* drow  = dn + (size_t)j * (LD * DIM) + (size_t)(wave * 16 + lo) * DIM;

    // Two accumulators per tile to break the WMMA D->C dependency chain.
    v8f c0a = {}, c0b = {}, c1a = {}, c1b = {};

#pragma unroll
    for (int k = 0; k < DIM; k += 8) {
        int offA = k + 2 * hi;
        int offB = k + 4 + 2 * hi;

        v2f b0 = *(const v2f*)(drow  + offA);
        v2f a00 = *(const v2f*)(qrow0 + offA);
        v2f a10 = *(const v2f*)(qrow1 + offA);
        c0a = __builtin_amdgcn_wmma_f32_16x16x4_f32(false, a00, false, b0,
                                                    (short)0, c0a, false, false);
        c1a = __builtin_amdgcn_wmma_f32_16x16x4_f32(false, a10, false, b0,
                                                    (short)0, c1a, false, false);

        v2f b1 = *(const v2f*)(drow  + offB);
        v2f a01 = *(const v2f*)(qrow0 + offB);
        v2f a11 = *(const v2f*)(qrow1 + offB);
        c0b = __builtin_amdgcn_wmma_f32_16x16x4_f32(false, a01, false, b1,
                                                    (short)0, c0b, false, false);
        c1b = __builtin_amdgcn_wmma_f32_16x16x4_f32(false, a11, false, b1,
                                                    (short)0, c1b, false, false);
    }

    // Per-lane max over the 16 accumulated sums (8 per tile, two tiles).
    float m = -3.0e38f;
#pragma unroll
    for (int v = 0; v < 8; ++v) {
        m = fmaxf(m, c0a[v] + c0b[v]);
        m = fmaxf(m, c1a[v] + c1b[v]);
    }
    // Wave max (covers all N columns and both M halves of the tiles).
#pragma unroll
    for (int s = 16; s >= 1; s >>= 1) m = fmaxf(m, __shfl_xor(m, s));

    __shared__ float red[16];
    if (lane == 0) red[wave] = m;
    __syncthreads();
    if (threadIdx.x == 0) {
        float t = red[0];
#pragma unroll
        for (int w = 1; w < 16; ++w) t = fmaxf(t, red[w]);
        simmax[i * BATCH + j] = t;
    }
}

// ---------------------------------------------------------------------------
// Kernel 3: hinge loss over sim[64][64] + abnormal-loss fallback.
// ---------------------------------------------------------------------------
__global__ __launch_bounds__(64) void finalize_kernel(const float* __restrict__ simmax,
                                                      float* __restrict__ out) {
    int i = threadIdx.x;  // 0..63
    float pos = simmax[i * BATCH + i];
    float s = 0.0f;
    for (int j = 0; j < BATCH; ++j) {
        if (j == i) continue;
        float h = MARGIN - pos + simmax[i * BATCH + j];
        s += fmaxf(h, 0.0f);
    }
    s *= (1.0f / (float)(BATCH - 1));

    __shared__ float red[BATCH];
    red[i] = s;
    __syncthreads();
    if (i == 0) {
        float t = 0.0f;
        for (int k = 0; k < BATCH; ++k) t += red[k];
        float loss = t * (1.0f / (float)BATCH);
        if (isnan(loss) || isinf(loss) || loss == 0.0f) loss = 1.0f;
        out[0] = loss;
    }
}

// ---------------------------------------------------------------------------
extern "C" void kernel_launch(void* const* d_in, const int* in_sizes, int n_in,
                              void* d_out, int out_size, void* d_ws, size_t ws_size,
                              hipStream_t stream) {
    const float* q = (const float*)d_in[0];  // [64, 32, 128] f32
    const float* d = (const float*)d_in[1];  // [64, 256, 128] f32
    float* out = (float*)d_out;              // scalar f32

    char* ws = (char*)d_ws;
    float* qn     = (float*)(ws);                       // 2048*128*4  = 1 MB
    float* dn     = (float*)(ws + (size_t)1048576);     // 16384*128*4 = 8 MB
    float* simmax = (float*)(ws + (size_t)9437184);     // 64*64*4     = 16 KB

    // 1) normalize: 18432 rows, one wave per row, 8 waves per block.
    l2norm_kernel<<<(BATCH * LQ + BATCH * LD) / 8, 256, 0, stream>>>(q, d, qn, dn);

    // 2) maxsim over all (i,j) batch pairs.
    dim3 grid(BATCH, BATCH, 1);
    maxsim_kernel<<<grid, 512, 0, stream>>>(qn, dn, simmax);

    // 3) loss.
    finalize_kernel<<<1, BATCH, 0, stream>>>(simmax, out);
}